// PredEnvHatYSep_8349416424170
// MI455X (gfx1250) — compile-verified
//
#include <hip/hip_runtime.h>

// MI455X / gfx1250, wave32. Dual rank-1 MLP (H=128) + select, built entirely on
// V_WMMA_F32_16X16X4_F32:
//   layer1: D1 = A(w1|b1, 16x4) x B(x|1, 4x16)      (bias fused via K=1 column)
//   relu  : v_max on D1 accumulator registers
//   layer2: acc += A(w2 bcast, 16x4) x B(D1 regs)   (D-layout == B-layout trick)
// Each wave handles 16 elements per iteration (N dimension of the WMMA tile).

typedef __attribute__((ext_vector_type(2))) float v2f;
typedef __attribute__((ext_vector_type(8))) float v8f;

#define NELEM 2097152
#define HDIM  128
#define NTILE (NELEM / 16)

__global__ __launch_bounds__(256) void PredEnvHatYSep_kernel(
    const int*   __restrict__ g,
    const float* __restrict__ x,
    const float* __restrict__ w1a, const float* __restrict__ b1a,
    const float* __restrict__ w2a, const float* __restrict__ b2a1,
    const float* __restrict__ w1b, const float* __restrict__ b1b,
    const float* __restrict__ w2b, const float* __restrict__ b2b1,
    float*       __restrict__ out)
{
  const int  lane = threadIdx.x & 31;
  const bool lo   = lane < 16;     // lane half: WMMA M/K slot selector
  const int  li   = lane & 15;

  const float* W1[2] = { w1a, w1b };
  const float* B1[2] = { b1a, b1b };
  const float* W2[2] = { w2a, w2b };

  // ---- loop-invariant A operands (weights are shared across all N) ----
  // layer1 A (16x4): lanes0-15 = (K0: w1[16t+lane], K1: b1[16t+lane]); lanes16-31 (K2,K3) = 0
  v2f a1[2][8];
  // layer2 A (16x4): broadcast w2 into K slots matching D1's register layout.
  // chunk c covers hidden rows {16t+2c, 16t+2c+1, 16t+2c+8, 16t+2c+9}:
  //   v.x: lanes0-15 = w2[16t+2c]   (K0), lanes16-31 = w2[16t+2c+8] (K2)
  //   v.y: lanes0-15 = w2[16t+2c+1] (K1), lanes16-31 = w2[16t+2c+9] (K3)
  v2f a2[2][8][4];

#pragma unroll
  for (int m = 0; m < 2; ++m) {
#pragma unroll
    for (int t = 0; t < 8; ++t) {
      float wv = W1[m][t * 16 + li];
      float bv = B1[m][t * 16 + li];
      a1[m][t].x = lo ? wv : 0.0f;
      a1[m][t].y = lo ? bv : 0.0f;
#pragma unroll
      for (int c = 0; c < 4; ++c) {
        int base = t * 16 + 2 * c + (lo ? 0 : 8);
        a2[m][t][c].x = W2[m][base];
        a2[m][t][c].y = W2[m][base + 1];
      }
    }
  }
  const float bias2_0 = b2a1[0];
  const float bias2_1 = b2b1[0];

  const int wavesPerBlock = blockDim.x >> 5;
  const int wave   = blockIdx.x * wavesPerBlock + (threadIdx.x >> 5);
  const int nWaves = gridDim.x * wavesPerBlock;

  for (int tile = wave; tile < NTILE; tile += nWaves) {
    const int ebase = tile * 16;

    // B operand for layer1: row K0 = x values (lanes0-15), K1 = ones, K2/K3 = 0
    float xv = x[ebase + li];
    v2f bx;
    bx.x = lo ? xv : 0.0f;
    bx.y = lo ? 1.0f : 0.0f;

    // stream the next tile of x through the gfx1250 prefetch path
    {
      int pf = ebase + nWaves * 16;
      if (pf < NELEM) __builtin_prefetch(x + pf + li, 0, 0);
    }

    v8f acc0 = {0.f, 0.f, 0.f, 0.f, 0.f, 0.f, 0.f, 0.f};
    v8f acc1 = {0.f, 0.f, 0.f, 0.f, 0.f, 0.f, 0.f, 0.f};

#pragma unroll
    for (int m = 0; m < 2; ++m) {
#pragma unroll
      for (int t = 0; t < 8; ++t) {
        v8f zero = {0.f, 0.f, 0.f, 0.f, 0.f, 0.f, 0.f, 0.f};
        // D1[m_h, n] = w1[16t+m_h]*x[n] + b1[16t+m_h]
        v8f d1 = __builtin_amdgcn_wmma_f32_16x16x4_f32(
            false, a1[m][t], false, bx, (short)0, zero, false, false);
#pragma unroll
        for (int r = 0; r < 8; ++r) d1[r] = fmaxf(d1[r], 0.0f);
        // layer2: D1 register pairs are directly legal B operands
#pragma unroll
        for (int c = 0; c < 4; ++c) {
          v2f hb;
          hb.x = d1[2 * c];
          hb.y = d1[2 * c + 1];
          if (m == 0)
            acc0 = __builtin_amdgcn_wmma_f32_16x16x4_f32(
                false, a2[m][t][c], false, hb, (short)0, acc0, false, false);
          else
            acc1 = __builtin_amdgcn_wmma_f32_16x16x4_f32(
                false, a2[m][t][c], false, hb, (short)0, acc1, false, false);
        }
      }
    }

    // every row of acc equals the output vector; row M=0 lives in VGPR0, lanes0-15
    float o1 = acc0[0] + bias2_0;
    float o2 = acc1[0] + bias2_1;
    int gv   = g[ebase + li];
    float res = (gv == 0) ? o1 : ((gv == 1) ? o2 : 0.0f);
    if (lo) out[ebase + lane] = res;
  }
}

extern "C" void kernel_launch(void* const* d_in, const int* in_sizes, int n_in,
                              void* d_out, int out_size, void* d_ws, size_t ws_size,
                              hipStream_t stream) {
  (void)in_sizes; (void)n_in; (void)d_ws; (void)ws_size; (void)out_size;
  const int*   g    = (const int*)d_in[0];
  const float* x    = (const float*)d_in[1];
  const float* w1_1 = (const float*)d_in[2];  // [H,1] -> H floats
  const float* b1_1 = (const float*)d_in[3];  // [H]
  const float* w1_2 = (const float*)d_in[4];  // [1,H] -> H floats
  const float* b1_2 = (const float*)d_in[5];  // [1]
  const float* w2_1 = (const float*)d_in[6];
  const float* b2_1 = (const float*)d_in[7];
  const float* w2_2 = (const float*)d_in[8];
  const float* b2_2 = (const float*)d_in[9];
  float* outp = (float*)d_out;

  dim3 block(256);           // 8 waves/block
  dim3 grid(2048);           // 16384 waves -> 8 tiles of 16 elements per wave
  PredEnvHatYSep_kernel<<<grid, block, 0, stream>>>(
      g, x,
      w1_1, b1_1, w1_2, b1_2,   // MLP branch 1: (w1,b1,w2,b2)
      w2_1, b2_1, w2_2, b2_2,   // MLP branch 2
      outp);
}